// S4_6957847020307
// MI455X (gfx1250) — compile-verified
//
#include <hip/hip_runtime.h>
#include <math.h>

#define H_DIM   512
#define N_STATE 64
#define L_SEQ_  2048
#define BATCH_  8
#define LF      4096
#define LN_EPS  1e-5f
#define PI_F    3.14159265358979323846f

typedef float v2f __attribute__((ext_vector_type(2)));
typedef float v8f __attribute__((ext_vector_type(8)));

// ---------------------------------------------------------------------------
// complex helpers (float2 = {re, im})
// ---------------------------------------------------------------------------
__device__ __forceinline__ float2 cmul(float2 a, float2 b) {
    return make_float2(a.x * b.x - a.y * b.y, a.x * b.y + a.y * b.x);
}
__device__ __forceinline__ float2 cadd(float2 a, float2 b) {
    return make_float2(a.x + b.x, a.y + b.y);
}
__device__ __forceinline__ float2 csub(float2 a, float2 b) {
    return make_float2(a.x - b.x, a.y - b.y);
}
__device__ __forceinline__ float2 cinv(float2 a) {
    float r = 1.0f / (a.x * a.x + a.y * a.y);
    return make_float2(a.x * r, -a.y * r);
}
__device__ __forceinline__ float2 cdiv(float2 a, float2 b) {
    return cmul(a, cinv(b));
}

// ---------------------------------------------------------------------------
// WMMA wrapper: D = A(16x4,f32) * B(4x16,f32) + C(16x16,f32)
// (neg_a, A, neg_b, B, c_mod, C, reuse_a, reuse_b)
// ---------------------------------------------------------------------------
__device__ __forceinline__ v8f wmma4(v2f a, v2f b, v8f c) {
    return __builtin_amdgcn_wmma_f32_16x16x4_f32(false, a, false, b,
                                                 (short)0, c, false, false);
}

// ---------------------------------------------------------------------------
// Kernel 1: K_f[h, l]  (Cauchy kernel evaluation), one block per h
// ---------------------------------------------------------------------------
__global__ __launch_bounds__(256) void kf_kernel(
    const float2* __restrict__ Lam, const float2* __restrict__ Pv,
    const float2* __restrict__ Bv,  const float2* __restrict__ Cm,
    const float*  __restrict__ log_step, float2* __restrict__ Kf)
{
    __shared__ float2 s_lam[N_STATE];
    __shared__ float2 s_w0[N_STATE], s_w1[N_STATE], s_w2[N_STATE], s_w3[N_STATE];

    const int h   = blockIdx.x;
    const int tid = threadIdx.x;

    if (tid < N_STATE) {
        float2 lam = Lam[tid];
        float2 c   = Cm[h * N_STATE + tid];
        float2 b   = Bv[tid];
        float2 p   = Pv[tid];
        float2 pc  = make_float2(p.x, -p.y);      // conj(P)
        s_lam[tid] = lam;
        s_w0[tid]  = cmul(c, b);                  // C*B
        s_w1[tid]  = cmul(c, p);                  // C*P
        s_w2[tid]  = cmul(pc, b);                 // conj(P)*B
        s_w3[tid]  = cmul(pc, p);                 // conj(P)*P
    }
    __syncthreads();

    const float sfac = 2.0f * __expf(-log_step[h]);   // 2/dt

    for (int l = tid; l < LF; l += 256) {
        float ang = -2.0f * PI_F * ((float)l / (float)LF);
        float sn, cs;
        __sincosf(ang, &sn, &cs);
        float2 z   = make_float2(0.999f * cs, 0.999f * sn);
        float2 num = make_float2(1.0f - z.x, -z.y);
        float2 den = make_float2(1.0f + z.x + 1e-6f, z.y);
        float2 zr  = cdiv(num, den);
        float2 s   = make_float2(sfac * zr.x, sfac * zr.y);

        float2 a0 = make_float2(0.f, 0.f), a1 = a0, a2 = a0, a3 = a0;
        #pragma unroll 4
        for (int n = 0; n < N_STATE; ++n) {
            float2 d = make_float2(s.x - s_lam[n].x, s.y - s_lam[n].y);
            float2 r = cinv(d);
            a0 = cadd(a0, cmul(s_w0[n], r));
            a1 = cadd(a1, cmul(s_w1[n], r));
            a2 = cadd(a2, cmul(s_w2[n], r));
            a3 = cadd(a3, cmul(s_w3[n], r));
        }
        float2 onek3 = make_float2(1.0f + a3.x, a3.y);
        float2 Ks    = csub(a0, cdiv(cmul(a1, a2), onek3));
        float2 onez  = make_float2(1.0f + z.x, z.y);
        float2 g     = cdiv(make_float2(2.0f, 0.0f), onez);
        Kf[(size_t)h * LF + l] = cmul(Ks, g);
    }
}

// ---------------------------------------------------------------------------
// One 16-column j-tile of a radix-16 Stockham stage (4096-pt FFT),
// 16-point DFT done as a complex matmul on V_WMMA_F32_16X16X4_F32.
//   gather:  v[a][j] = src[j + a*256],  j = jt*16 + (lane&15)
//   twiddle: v[a][j] *= exp(-2*pi*i * a * (j%Ns) / (16*Ns))
//   DFT16:   d[k][j] = sum_a F16[k][a] * v[a][j]
//   scatter: dst[(j/Ns)*16*Ns + (j%Ns) + k*Ns] = d[k][j]
// No control flow: EXEC is all-ones around every WMMA.
// ---------------------------------------------------------------------------
__device__ __forceinline__ void fft_stage_tile(
    const float2* __restrict__ src, float2* __restrict__ dst,
    int log2Ns, int jt, int lane,
    const v2f Fr[4], const v2f Fi[4], const v2f Fin[4])
{
    const int   n    = lane & 15;
    const int   half = lane >> 4;                       // lane group 0/1
    const int   Ns   = 1 << log2Ns;
    const float twf  = -2.0f * PI_F / (float)(16 << log2Ns);

    const int j       = jt * 16 + n;                    // my column
    const int jm      = j & (Ns - 1);                   // j % Ns
    const int idxbase = ((j >> log2Ns) << (log2Ns + 4)) + jm;

    v8f dr = {0.f,0.f,0.f,0.f,0.f,0.f,0.f,0.f};
    v8f di = {0.f,0.f,0.f,0.f,0.f,0.f,0.f,0.f};

    #pragma unroll
    for (int kk = 0; kk < 4; ++kk) {
        const int r0 = kk * 4 + half * 2;               // B rows this lane feeds
        float2 x0 = src[j + r0 * 256];
        float2 x1 = src[j + (r0 + 1) * 256];
        // inter-stage twiddles
        float sA, cA, sB, cB;
        __sincosf(twf * (float)(r0 * jm), &sA, &cA);
        __sincosf(twf * (float)((r0 + 1) * jm), &sB, &cB);
        float2 t0 = make_float2(x0.x * cA - x0.y * sA, x0.x * sA + x0.y * cA);
        float2 t1 = make_float2(x1.x * cB - x1.y * sB, x1.x * sB + x1.y * cB);
        v2f br = {t0.x, t1.x};
        v2f bi = {t0.y, t1.y};
        // complex matmul: Dr += Fr*Vr - Fi*Vi ; Di += Fr*Vi + Fi*Vr
        dr = wmma4(Fr[kk],  br, dr);
        dr = wmma4(Fin[kk], bi, dr);
        di = wmma4(Fr[kk],  bi, di);
        di = wmma4(Fi[kk],  br, di);
    }
    #pragma unroll
    for (int vg = 0; vg < 8; ++vg) {
        int m = vg + half * 8;                          // D-matrix row
        dst[idxbase + m * Ns] = make_float2(dr[vg], di[vg]);
    }
}

// full stage: 8 waves cover 16 tiles, two straight-line calls per wave
__device__ __forceinline__ void fft_stage(
    const float2* __restrict__ src, float2* __restrict__ dst,
    int log2Ns, int wave, int lane,
    const v2f Fr[4], const v2f Fi[4], const v2f Fin[4])
{
    fft_stage_tile(src, dst, log2Ns, wave,     lane, Fr, Fi, Fin);
    fft_stage_tile(src, dst, log2Ns, wave + 8, lane, Fr, Fi, Fin);
}

// ---------------------------------------------------------------------------
// Kernel 2: per-(b,h) sequence: FFT(4096) -> *K_f -> IFFT -> +D*u -> GELU
// ---------------------------------------------------------------------------
__global__ __launch_bounds__(256) void s4_fft_kernel(
    const float*  __restrict__ u, const float2* __restrict__ Kf,
    const float*  __restrict__ Dv, float* __restrict__ y_out)
{
    __shared__ float2 bufA[LF];
    __shared__ float2 bufB[LF];

    const int bid  = blockIdx.x;        // b*H + h
    const int h    = bid & (H_DIM - 1);
    const int tid  = threadIdx.x;
    const int lane = tid & 31;
    const int wave = tid >> 5;
    const int half = lane >> 4;
    const int n    = lane & 15;

    const float*  urow = u  + (size_t)bid * L_SEQ_;
    const float2* Krow = Kf + (size_t)h   * LF;
    __builtin_prefetch(urow + tid * 8, 0, 0);           // global_prefetch_b8
    __builtin_prefetch(Krow + tid * 16, 0, 0);

    // constant DFT-16 A-fragments: F[m][a] = exp(-2*pi*i*m*a/16)
    v2f Fr[4], Fi[4], Fin[4];
    {
        const float mf = (float)n;
        #pragma unroll
        for (int kk = 0; kk < 4; ++kk) {
            const int c0 = kk * 4 + half * 2;
            float aA = -(2.0f * PI_F / 16.0f) * mf * (float)c0;
            float aB = -(2.0f * PI_F / 16.0f) * mf * (float)(c0 + 1);
            float s0, cc0, s1, cc1;
            __sincosf(aA, &s0, &cc0);
            __sincosf(aB, &s1, &cc1);
            Fr[kk]  = { cc0,  cc1};
            Fi[kk]  = { s0,   s1 };
            Fin[kk] = {-s0,  -s1 };
        }
    }

    for (int t = tid; t < L_SEQ_; t += 256) {
        bufA[t]          = make_float2(urow[t], 0.0f);
        bufA[t + L_SEQ_] = make_float2(0.0f, 0.0f);
    }
    __syncthreads();

    // forward FFT: A->B (Ns=1), B->A (Ns=16), A->B (Ns=256); sorted in bufB
    fft_stage(bufA, bufB, 0, wave, lane, Fr, Fi, Fin); __syncthreads();
    fft_stage(bufB, bufA, 4, wave, lane, Fr, Fi, Fin); __syncthreads();
    fft_stage(bufA, bufB, 8, wave, lane, Fr, Fi, Fin); __syncthreads();

    // pointwise: conj(U_f * K_f) for inverse-via-forward trick
    for (int l = tid; l < LF; l += 256) {
        float2 p = cmul(bufB[l], Krow[l]);
        bufA[l] = make_float2(p.x, -p.y);
    }
    __syncthreads();

    fft_stage(bufA, bufB, 0, wave, lane, Fr, Fi, Fin); __syncthreads();
    fft_stage(bufB, bufA, 4, wave, lane, Fr, Fi, Fin); __syncthreads();
    fft_stage(bufA, bufB, 8, wave, lane, Fr, Fi, Fin); __syncthreads();

    // y = real(ifft)[0:L] + D*u, then exact-erf GELU
    const float dh = Dv[h];
    float* yrow = y_out + (size_t)bid * L_SEQ_;
    for (int t = tid; t < L_SEQ_; t += 256) {
        float yv = bufB[t].x * (1.0f / (float)LF);   // conj -> same real part
        yv += dh * urow[t];
        yrow[t] = 0.5f * yv * (1.0f + erff(yv * 0.70710678118654752440f));
    }
}

// ---------------------------------------------------------------------------
// Kernel 3: LayerNorm over H at each (b, l); one block per position
// ---------------------------------------------------------------------------
__global__ __launch_bounds__(256) void ln_kernel(
    const float* __restrict__ y, const float* __restrict__ gamma,
    const float* __restrict__ beta, float* __restrict__ out)
{
    __shared__ float s_sum[256];
    __shared__ float s_sq[256];

    const int bid = blockIdx.x;                 // b*L + l
    const int b   = bid >> 11;                  // / 2048
    const int l   = bid & (L_SEQ_ - 1);
    const int tid = threadIdx.x;

    const size_t base = ((size_t)b * H_DIM) * L_SEQ_ + (size_t)l;
    const float v0 = y[base + (size_t)tid * L_SEQ_];
    const float v1 = y[base + (size_t)(tid + 256) * L_SEQ_];

    s_sum[tid] = v0 + v1;
    s_sq[tid]  = v0 * v0 + v1 * v1;
    __syncthreads();
    for (int off = 128; off > 0; off >>= 1) {
        if (tid < off) {
            s_sum[tid] += s_sum[tid + off];
            s_sq[tid]  += s_sq[tid + off];
        }
        __syncthreads();
    }
    const float mean = s_sum[0] * (1.0f / (float)H_DIM);
    const float var  = s_sq[0] * (1.0f / (float)H_DIM) - mean * mean;
    const float rstd = rsqrtf(var + LN_EPS);

    out[base + (size_t)tid * L_SEQ_] =
        (v0 - mean) * rstd * gamma[tid] + beta[tid];
    out[base + (size_t)(tid + 256) * L_SEQ_] =
        (v1 - mean) * rstd * gamma[tid + 256] + beta[tid + 256];
}

// ---------------------------------------------------------------------------
// launch
// ---------------------------------------------------------------------------
extern "C" void kernel_launch(void* const* d_in, const int* in_sizes, int n_in,
                              void* d_out, int out_size, void* d_ws, size_t ws_size,
                              hipStream_t stream)
{
    const float*  u        = (const float*) d_in[0];   // (B,H,L) f32
    const float2* Lam      = (const float2*)d_in[1];   // (N,)  c64
    const float2* Pv       = (const float2*)d_in[2];   // (N,)  c64
    const float2* Bv       = (const float2*)d_in[3];   // (N,)  c64
    const float2* Cm       = (const float2*)d_in[4];   // (H,N) c64
    const float*  log_step = (const float*) d_in[5];   // (H,)
    const float*  Dv       = (const float*) d_in[6];   // (H,)
    const float*  gamma    = (const float*) d_in[7];   // (H,)
    const float*  beta     = (const float*) d_in[8];   // (H,)
    float* out = (float*)d_out;

    char*   ws = (char*)d_ws;
    float2* Kf = (float2*)ws;                                        // 16 MB
    float*  yb = (float*)(ws + (size_t)H_DIM * LF * sizeof(float2)); // 33.5 MB

    kf_kernel    <<<H_DIM,           256, 0, stream>>>(Lam, Pv, Bv, Cm, log_step, Kf);
    s4_fft_kernel<<<BATCH_ * H_DIM,  256, 0, stream>>>(u, Kf, Dv, yb);
    ln_kernel    <<<BATCH_ * L_SEQ_, 256, 0, stream>>>(yb, gamma, beta, out);
}